// Decoder_9663676416359
// MI455X (gfx1250) — compile-verified
//
#include <hip/hip_runtime.h>
#include <hip/hip_bf16.h>
#include <cmath>

#define DIM_EMB   1024
#define DIM_HID   1024
#define NUM_VOCAB 32000
#define MAX_STEPS 101
#define NPART     500   // 32000 / (4 waves * 16 rows)

#define KCHUNK    64                // k-columns per TDM tile
#define LDS_PITCH (KCHUNK + 2)      // +2 DWORD pad per row (TDM pad_amount)
#define TILE_F    (64 * LDS_PITCH)  // 64 rows per block tile

typedef float v2f __attribute__((ext_vector_type(2)));
typedef float v8f __attribute__((ext_vector_type(8)));
typedef unsigned int u32x4 __attribute__((ext_vector_type(4)));
typedef int i32x4 __attribute__((ext_vector_type(4)));
typedef int i32x8 __attribute__((ext_vector_type(8)));

#if defined(__gfx1250__) && __has_builtin(__builtin_amdgcn_wmma_f32_16x16x4_f32)
#define USE_WMMA4 1
#else
#define USE_WMMA4 0
#endif

#if USE_WMMA4 && __has_builtin(__builtin_amdgcn_tensor_load_to_lds) && \
    __has_builtin(__builtin_amdgcn_s_wait_tensorcnt)
#define USE_TDM 1
#else
#define USE_TDM 0
#endif

__device__ __forceinline__ v8f wmma4(v2f a, v2f b, v8f c)
{
#if USE_WMMA4
    return __builtin_amdgcn_wmma_f32_16x16x4_f32(false, a, false, b,
                                                 (short)0, c, false, false);
#else
    return c;
#endif
}

#if USE_TDM
__device__ __forceinline__ unsigned lds_off(const void* p)
{
    // Low 32 bits of a generic pointer into LDS are the LDS byte address
    // (flat aperture mapping, ISA 10.2).
    return (unsigned)(unsigned long long)(uintptr_t)p;
}

// TDM: DMA a 64-row x KCHUNK-col fp32 tile of W (row stride = rowStrideElems)
// into LDS at lds_byte, inserting 2 pad DWORDs after every row (pitch 66).
__device__ __forceinline__ void tdm_load_tile(const float* gsrc,
                                              unsigned lds_byte,
                                              unsigned rowStrideElems)
{
    unsigned long long ga = (unsigned long long)(uintptr_t)gsrc;
    u32x4 g0;
    g0[0] = 1u;                                      // count=1, user descriptor
    g0[1] = lds_byte;                                // D#.lds_addr
    g0[2] = (unsigned)ga;                            // global_addr[31:0]
    g0[3] = (unsigned)((ga >> 32) & 0x1FFFFFFull)    // global_addr[56:32]
          | (2u << 30);                              // type = 2 ("image")
    i32x8 g1;
    g1[0] = (2 << 16)        // data_size = 4 bytes
          | (1 << 20)        // pad_enable
          | (5 << 22)        // pad_interval: 2^(5+1) = 64 DWORDs (one row)
          | (1 << 25);       // pad_amount: 2 DWORDs
    g1[1] = 0;                           // barrier addr=0 | tensor_dim0[15:0]=0
    g1[2] = 0x10;                        // tensor_dim0=0x100000 (never OOB) | dim1[15:0]=0
    g1[3] = 0x10 | (KCHUNK << 16);       // tensor_dim1=0x100000 | tile_dim0=KCHUNK
    g1[4] = 64;                          // tile_dim1 = 64 rows, tile_dim2 = 0
    g1[5] = (int)rowStrideElems;         // tensor_dim0_stride[31:0]
    g1[6] = 0;
    g1[7] = 0;
    i32x4 gz = {0, 0, 0, 0};
#if __clang_major__ >= 23
    i32x8 gz8 = {0, 0, 0, 0, 0, 0, 0, 0};
    __builtin_amdgcn_tensor_load_to_lds(g0, g1, gz, gz, gz8, 0);
#else
    __builtin_amdgcn_tensor_load_to_lds(g0, g1, gz, gz, 0);
#endif
}

// Block-cooperative GEMV: 64 rows per block (4 waves x 16 rows), weights
// staged through LDS by TDM with double buffering; V_WMMA_F32_16X16X4_F32
// consumes A from LDS, B = x replicated across the 16 N columns.
__device__ __forceinline__ v8f gemv_tdm_block(const float* __restrict__ Wblk,
                                              const float* __restrict__ x,
                                              int K, float* t0, float* t1,
                                              int wave, int half, int m, v8f acc)
{
    const int nch = K / KCHUNK;
    if (threadIdx.x < 32)
        tdm_load_tile(Wblk, lds_off(t0), (unsigned)K);
    for (int c = 0; c < nch; ++c) {
        float* tb = (c & 1) ? t1 : t0;
        if (threadIdx.x < 32)
            __builtin_amdgcn_s_wait_tensorcnt(0);
        __syncthreads();                              // tile[c] visible to all
        if ((c + 1) < nch && threadIdx.x < 32)
            tdm_load_tile(Wblk + (size_t)(c + 1) * KCHUNK,
                          lds_off((c & 1) ? t0 : t1), (unsigned)K);
        const float* trow = tb + (wave * 16 + m) * LDS_PITCH + 2 * half;
        const float* xb   = x + c * KCHUNK + 2 * half;
        #pragma unroll 8
        for (int k = 0; k < KCHUNK; k += 4) {
            v2f a = *(const v2f*)(trow + k);   // A: lane(16h+m) -> W[m][k+2h+v]
            v2f b = *(const v2f*)(xb + k);     // B: x[k+2h+v] replicated over N
            acc = wmma4(a, b, acc);
        }
        __syncthreads();                              // done reading tile[c]
    }
    return acc;
}
#else
// Fallback: direct-global WMMA GEMV.
__device__ __forceinline__ v8f wave_gemv16(const float* __restrict__ W,
                                           int rowBase, int K,
                                           const float* __restrict__ x,
                                           int half, int m, v8f acc)
{
#if USE_WMMA4
    const float* wrow = W + (size_t)(rowBase + m) * (size_t)K + 2 * half;
    const float* xb   = x + 2 * half;
    #pragma unroll 8
    for (int k = 0; k < K; k += 4) {
        v2f a = *(const v2f*)(wrow + k);
        v2f b = *(const v2f*)(xb + k);
        acc = wmma4(a, b, acc);
    }
#else
    for (int r = 0; r < 8; ++r) {
        int row = rowBase + 8 * half + r;
        float s = 0.f;
        for (int k = 0; k < K; ++k)
            s = fmaf(W[(size_t)row * K + k], x[k], s);
        acc[r] += s;
    }
#endif
    return acc;
}
#endif  // USE_TDM

__device__ __forceinline__ float sigf(float v) { return 1.0f / (1.0f + expf(-v)); }

// ---------------- init: h from input, c = 0, tok = BOS -------------------
__global__ void init_state_kernel(const float* __restrict__ h_in,
                                  const int* __restrict__ id_bos,
                                  float* __restrict__ hbuf,
                                  float* __restrict__ cbuf,
                                  int* __restrict__ tok)
{
    int i = blockIdx.x * blockDim.x + threadIdx.x;
    if (i < DIM_HID) { hbuf[i] = h_in[i]; cbuf[i] = 0.0f; }
    if (i == 0) *tok = *id_bos;
}

// ---------------- gates = W_ih*emb[tok] + b_ih + W_hh*h + b_hh -----------
__global__ void lstm_gates_kernel(const float* __restrict__ emb,
                                  const float* __restrict__ W_ih,
                                  const float* __restrict__ W_hh,
                                  const float* __restrict__ b_ih,
                                  const float* __restrict__ b_hh,
                                  const float* __restrict__ hbuf,
                                  const int*   __restrict__ tok,
                                  float* __restrict__ gates)
{
#if USE_TDM
    __shared__ float t0[TILE_F];
    __shared__ float t1[TILE_F];
#endif
    const int lane  = threadIdx.x & 31;
    const int wave  = threadIdx.x >> 5;
    const int half  = lane >> 4;
    const int m     = lane & 15;
    const int rowBase = blockIdx.x * 64 + wave * 16;    // 4096 rows total

    const float* x = emb + (size_t)(*tok) * DIM_EMB;

    v8f acc = {};
#if USE_TDM
    acc = gemv_tdm_block(W_ih + (size_t)(blockIdx.x * 64) * DIM_EMB, x,
                         DIM_EMB, t0, t1, wave, half, m, acc);
    acc = gemv_tdm_block(W_hh + (size_t)(blockIdx.x * 64) * DIM_HID, hbuf,
                         DIM_HID, t0, t1, wave, half, m, acc);
#else
    acc = wave_gemv16(W_ih, rowBase, DIM_EMB, x,    half, m, acc);
    acc = wave_gemv16(W_hh, rowBase, DIM_HID, hbuf, half, m, acc);
#endif

    if (m == 0) {
        const int r0 = rowBase + 8 * half;
        #pragma unroll
        for (int r = 0; r < 8; ++r) {
            const int row = r0 + r;
            gates[row] = acc[r] + b_ih[row] + b_hh[row];
        }
    }
}

// ---------------- LSTM pointwise: c,h update -----------------------------
__global__ void lstm_pointwise_kernel(const float* __restrict__ gates,
                                      float* __restrict__ hbuf,
                                      float* __restrict__ cbuf)
{
    int i = blockIdx.x * blockDim.x + threadIdx.x;
    if (i >= DIM_HID) return;
    float ig = sigf(gates[i]);
    float fg = sigf(gates[DIM_HID + i]);
    float gg = tanhf(gates[2 * DIM_HID + i]);
    float og = sigf(gates[3 * DIM_HID + i]);
    float c  = fg * cbuf[i] + ig * gg;
    cbuf[i]  = c;
    hbuf[i]  = og * tanhf(c);
}

// ---------------- logits GEMV + per-block argmax partial -----------------
__global__ void logits_argmax_kernel(const float* __restrict__ W_lin,
                                     const float* __restrict__ b_lin,
                                     const float* __restrict__ hbuf,
                                     float* __restrict__ partVal,
                                     int*   __restrict__ partIdx)
{
#if USE_TDM
    __shared__ float t0[TILE_F];
    __shared__ float t1[TILE_F];
#endif
    const int lane  = threadIdx.x & 31;
    const int wave  = threadIdx.x >> 5;
    const int half  = lane >> 4;
    const int m     = lane & 15;
    const int rowBase = blockIdx.x * 64 + wave * 16;    // 32000 rows total

    v8f acc = {};
#if USE_TDM
    acc = gemv_tdm_block(W_lin + (size_t)(blockIdx.x * 64) * DIM_HID, hbuf,
                         DIM_HID, t0, t1, wave, half, m, acc);
#else
    acc = wave_gemv16(W_lin, rowBase, DIM_HID, hbuf, half, m, acc);
#endif

    __shared__ float sv[8];
    __shared__ int   si[8];

    if (m == 0) {
        const int r0 = rowBase + 8 * half;
        float bestV = acc[0] + b_lin[r0];
        int   bestI = r0;
        #pragma unroll
        for (int r = 1; r < 8; ++r) {
            float v = acc[r] + b_lin[r0 + r];
            if (v > bestV) { bestV = v; bestI = r0 + r; }   // strict >: first wins
        }
        const int slot = wave * 2 + half;  // ascending row-index order
        sv[slot] = bestV;
        si[slot] = bestI;
    }
    __syncthreads();
    if (threadIdx.x == 0) {
        float bestV = sv[0];
        int   bestI = si[0];
        for (int s = 1; s < 8; ++s)
            if (sv[s] > bestV) { bestV = sv[s]; bestI = si[s]; }
        partVal[blockIdx.x] = bestV;
        partIdx[blockIdx.x] = bestI;
    }
}

// ---------------- final argmax over partials, emit token -----------------
__global__ void argmax_final_kernel(const float* __restrict__ partVal,
                                    const int*   __restrict__ partIdx,
                                    int* __restrict__ tok,
                                    int* __restrict__ out_ids,
                                    int step)
{
    __shared__ float sv[512];
    __shared__ int   si[512];
    const int t = threadIdx.x;

    float bestV = -INFINITY;
    int   bestI = 0x7fffffff;
    for (int p = t; p < NPART; p += blockDim.x) {
        float v = partVal[p];
        int   i = partIdx[p];
        if (v > bestV || (v == bestV && i < bestI)) { bestV = v; bestI = i; }
    }
    sv[t] = bestV; si[t] = bestI;
    __syncthreads();
    for (int s = blockDim.x >> 1; s > 0; s >>= 1) {
        if (t < s) {
            if (sv[t + s] > sv[t] ||
                (sv[t + s] == sv[t] && si[t + s] < si[t])) {
                sv[t] = sv[t + s]; si[t] = si[t + s];
            }
        }
        __syncthreads();
    }
    if (t == 0) {
        *tok = si[0];
        out_ids[step] = si[0];
    }
}

// =========================================================================
extern "C" void kernel_launch(void* const* d_in, const int* in_sizes, int n_in,
                              void* d_out, int out_size, void* d_ws, size_t ws_size,
                              hipStream_t stream)
{
    const float* h_in   = (const float*)d_in[0];   // [1,1,1024]
    const float* emb    = (const float*)d_in[1];   // [32000,1024]
    const float* W_ih   = (const float*)d_in[2];   // [4096,1024]
    const float* W_hh   = (const float*)d_in[3];   // [4096,1024]
    const float* b_ih   = (const float*)d_in[4];   // [4096]
    const float* b_hh   = (const float*)d_in[5];   // [4096]
    const float* W_lin  = (const float*)d_in[6];   // [32000,1024]
    const float* b_lin  = (const float*)d_in[7];   // [32000]
    const int*   id_bos = (const int*)d_in[8];
    // d_in[9] = id_eos: unused (reference runs fixed 101 steps)

    int* out_ids = (int*)d_out;

    // workspace layout (bytes)
    char*  ws      = (char*)d_ws;
    float* hbuf    = (float*)(ws + 0);        // 1024 f
    float* cbuf    = (float*)(ws + 4096);     // 1024 f
    float* gates   = (float*)(ws + 8192);     // 4096 f
    float* partVal = (float*)(ws + 24576);    // 500 f
    int*   partIdx = (int*)  (ws + 26624);    // 500 i
    int*   tok     = (int*)  (ws + 28672);    // 1 i

    init_state_kernel<<<4, 256, 0, stream>>>(h_in, id_bos, hbuf, cbuf, tok);

    for (int step = 0; step < MAX_STEPS; ++step) {
        // 4096 rows / (4 waves * 16 rows) = 64 blocks
        lstm_gates_kernel<<<64, 128, 0, stream>>>(emb, W_ih, W_hh, b_ih, b_hh,
                                                  hbuf, tok, gates);
        lstm_pointwise_kernel<<<4, 256, 0, stream>>>(gates, hbuf, cbuf);
        // 32000 rows / (4 waves * 16 rows) = 500 blocks
        logits_argmax_kernel<<<NPART, 128, 0, stream>>>(W_lin, b_lin, hbuf,
                                                        partVal, partIdx);
        argmax_final_kernel<<<1, 512, 0, stream>>>(partVal, partIdx, tok,
                                                   out_ids, step);
    }
}